// FUNKAN_22978075034121
// MI455X (gfx1250) — compile-verified
//
#include <hip/hip_runtime.h>
#include <hip/hip_bf16.h>
#include <math.h>

// Problem constants (from reference)
#define NB   16
#define NC   512
#define NHW  4096          // 64*64
#define NO   512
#define NR   6             // DEGREE+1
#define RP   8             // r padded to 8 for WMMA K
#define LN_EPS 1e-5f

// Workspace layout (floats)
#define PSI_OFF   0                      // [6][512]  raw psi, row-major
#define WPSI_OFF  (PSI_OFF + NR*NC)      // [512][8]  ln_w*psi, c-major, padded
#define MT_OFF    (WPSI_OFF + NC*RP)     // [512][8]  M[o][r] padded
#define S1_OFF    (MT_OFF + NO*RP)       // [8]
#define S2_OFF    (S1_OFF + 8)           // [8]

typedef __attribute__((ext_vector_type(2))) float v2f;
typedef __attribute__((ext_vector_type(8))) float v8f;

// ---------------------------------------------------------------------------
// Kernel 1: Hermite basis psi[r][c], w_psi[c][r]=ln_w*psi, S1[r]=sum(ln_w*psi),
// S2[r]=sum(ln_b*psi).  One block, 512 threads (c = tid).
// ---------------------------------------------------------------------------
__global__ __launch_bounds__(512) void hermite_setup_kernel(
    const float* __restrict__ ln_w, const float* __restrict__ ln_b,
    float* __restrict__ ws) {
  const int c = threadIdx.x;
  // x_d = tanh(linspace(-1,1,C)) * (sqrt(11)+1)
  const float xd = tanhf(-1.0f + 2.0f * (float)c / (float)(NC - 1)) * 4.31662479035539985f;
  const float g  = expf(-0.5f * xd * xd);
  const float pim4 = 0.75112554446494248f;   // pi^(-1/4)
  float p[NR];
  p[0] = pim4 * g;
  p[1] = 1.41421356237309515f * pim4 * xd * g;
#pragma unroll
  for (int k = 2; k < NR; ++k)
    p[k] = sqrtf(2.0f / (float)k) * xd * p[k - 1]
         - sqrtf((float)(k - 1) / (float)k) * p[k - 2];

  const float w  = ln_w[c];
  const float lb = ln_b[c];
#pragma unroll
  for (int r = 0; r < NR; ++r) {
    ws[PSI_OFF  + r * NC + c] = p[r];
    ws[WPSI_OFF + c * RP + r] = w * p[r];
  }
  ws[WPSI_OFF + c * RP + 6] = 0.0f;
  ws[WPSI_OFF + c * RP + 7] = 0.0f;

  // block reductions for S1, S2
  __shared__ float buf[NC];
#pragma unroll 1
  for (int r = 0; r < NR; ++r) {
    buf[c] = w * p[r];
    __syncthreads();
    for (int st = NC / 2; st > 0; st >>= 1) {
      if (c < st) buf[c] += buf[c + st];
      __syncthreads();
    }
    if (c == 0) ws[S1_OFF + r] = buf[0];
    __syncthreads();
    buf[c] = lb * p[r];
    __syncthreads();
    for (int st = NC / 2; st > 0; st >>= 1) {
      if (c < st) buf[c] += buf[c + st];
      __syncthreads();
    }
    if (c == 0) ws[S2_OFF + r] = buf[0];
    __syncthreads();
  }
  if (c == 0) { ws[S1_OFF + 6] = 0.f; ws[S1_OFF + 7] = 0.f;
                ws[S2_OFF + 6] = 0.f; ws[S2_OFF + 7] = 0.f; }
}

// ---------------------------------------------------------------------------
// Kernel 2: M[o][r] = sum_c psi[r][c] * fc_w[o][c]  (padded to 8 r-slots)
// ---------------------------------------------------------------------------
__global__ __launch_bounds__(256) void fold_fc_kernel(
    const float* __restrict__ fc_w, float* __restrict__ ws) {
  const int o = blockIdx.x * blockDim.x + threadIdx.x;
  if (o >= NO) return;
  float acc[NR] = {0.f, 0.f, 0.f, 0.f, 0.f, 0.f};
  const float* wr = fc_w + (size_t)o * NC;
#pragma unroll 4
  for (int c = 0; c < NC; ++c) {
    const float fw = wr[c];
#pragma unroll
    for (int r = 0; r < NR; ++r) acc[r] += ws[PSI_OFF + r * NC + c] * fw;
  }
#pragma unroll
  for (int r = 0; r < NR; ++r) ws[MT_OFF + o * RP + r] = acc[r];
  ws[MT_OFF + o * RP + 6] = 0.0f;
  ws[MT_OFF + o * RP + 7] = 0.0f;
}

// ---------------------------------------------------------------------------
// Kernel 3: fused LN-stats + scores + softmax + WMMA output GEMM.
// Grid: 256 blocks (16 batches x 16 position tiles), 256 threads (8 waves).
// Each thread owns one spatial position; each wave does the WMMA epilogue
// for its 32 positions (2 tiles of 16) x 32 o-tiles, computing out^T tiles
// with V_WMMA_F32_16X16X4_F32 (K = r, padded 6->8, two k-steps of 4).
// ---------------------------------------------------------------------------
__global__ __launch_bounds__(256) void fused_main_kernel(
    const float* __restrict__ x, const float* __restrict__ fc_b,
    const float* __restrict__ ws, float* __restrict__ out) {
  const int tid   = threadIdx.x;
  const int b     = blockIdx.x >> 4;
  const int pbase = (blockIdx.x & 15) * 256;

  __shared__ __align__(32) float attn_s[256][RP];   // 8 KB
  __shared__ __align__(32) float mt_s[NO][RP];      // 16 KB
  __shared__ __align__(16) float fcb_s[NO];         // 2 KB

  // cooperative LDS fill (M tile + bias)
  {
    const float4* src = (const float4*)(ws + MT_OFF);
    float4* dst = (float4*)&mt_s[0][0];
#pragma unroll
    for (int i = tid; i < NO * RP / 4; i += 256) dst[i] = src[i];
    for (int i = tid; i < NO; i += 256) fcb_s[i] = fc_b[i];
  }

  float S1[NR], S2[NR];
#pragma unroll
  for (int r = 0; r < NR; ++r) { S1[r] = ws[S1_OFF + r]; S2[r] = ws[S2_OFF + r]; }

  // ---- Phase 1: streaming reduction over channels (coalesced: lane = pos) --
  const float* xp = x + (size_t)b * NC * NHW + pbase + tid;
  const float* wp = ws + WPSI_OFF;                   // wave-uniform reads
  float s = 0.f, s2 = 0.f;
  float d[NR] = {0.f, 0.f, 0.f, 0.f, 0.f, 0.f};
#pragma unroll 4
  for (int c = 0; c < NC; ++c) {
    const float v = xp[(size_t)c * NHW];
    const float* wr = wp + c * RP;
    s += v; s2 += v * v;
#pragma unroll
    for (int r = 0; r < NR; ++r) d[r] += v * wr[r];
  }
  const float mu   = s * (1.0f / NC);
  const float var  = s2 * (1.0f / NC) - mu * mu;
  const float rstd = 1.0f / sqrtf(var + LN_EPS);

  // scores + softmax over r
  float sc[NR], mx = -3.4e38f;
#pragma unroll
  for (int r = 0; r < NR; ++r) {
    sc[r] = (rstd * (d[r] - mu * S1[r]) + S2[r]) * 0.044194173824159216f; // 1/sqrt(512)
    mx = fmaxf(mx, sc[r]);
  }
  float se = 0.f;
#pragma unroll
  for (int r = 0; r < NR; ++r) { sc[r] = expf(sc[r] - mx); se += sc[r]; }
  const float inv = 1.0f / se;
#pragma unroll
  for (int r = 0; r < NR; ++r) attn_s[tid][r] = sc[r] * inv;
  attn_s[tid][6] = 0.0f;
  attn_s[tid][7] = 0.0f;
  __syncthreads();

  // ---- Phase 3: out^T tiles via f32 WMMA -----------------------------------
  // A (16x4) rows = o, B (4x16) cols = positions, D rows = o / cols = pos.
  const int lane = tid & 31;
  const int wv   = tid >> 5;
  const int half = lane >> 4;      // K-pair selector per ISA A/B layout
  const int l16  = lane & 15;
  const size_t ob = (size_t)b * NO * NHW + (size_t)pbase;

#pragma unroll 1
  for (int pt = 0; pt < 2; ++pt) {
    const int prow = wv * 32 + pt * 16 + l16;        // this lane's B column
    const v2f b01 = *(const v2f*)&attn_s[prow][half * 2];       // K {0,1}/{2,3}
    const v2f b23 = *(const v2f*)&attn_s[prow][4 + half * 2];   // K {4,5}/{6,7}
#pragma unroll 4
    for (int ot = 0; ot < NO / 16; ++ot) {
      const int orow = ot * 16 + l16;
      const v2f a01 = *(const v2f*)&mt_s[orow][half * 2];
      const v2f a23 = *(const v2f*)&mt_s[orow][4 + half * 2];
      // accumulator pre-loaded with bias: VGPR j <-> o = ot*16 + half*8 + j
      v8f acc = *(const v8f*)&fcb_s[ot * 16 + half * 8];
      acc = __builtin_amdgcn_wmma_f32_16x16x4_f32(
          false, a01, false, b01, (short)0, acc, false, false);
      acc = __builtin_amdgcn_wmma_f32_16x16x4_f32(
          false, a23, false, b23, (short)0, acc, false, false);
      // store: per VGPR j, 16 consecutive positions per half-wave (64B segs)
      float* op = out + ob + (size_t)(ot * 16 + half * 8) * NHW
                      + (wv * 32 + pt * 16 + l16);
#pragma unroll
      for (int j = 0; j < 8; ++j) op[(size_t)j * NHW] = acc[j];
    }
  }
}

// ---------------------------------------------------------------------------
extern "C" void kernel_launch(void* const* d_in, const int* in_sizes, int n_in,
                              void* d_out, int out_size, void* d_ws, size_t ws_size,
                              hipStream_t stream) {
  const float* x    = (const float*)d_in[0];
  const float* ln_w = (const float*)d_in[1];
  const float* ln_b = (const float*)d_in[2];
  const float* fc_w = (const float*)d_in[3];
  const float* fc_b = (const float*)d_in[4];
  float* ws  = (float*)d_ws;
  float* out = (float*)d_out;

  hipLaunchKernelGGL(hermite_setup_kernel, dim3(1), dim3(512), 0, stream,
                     ln_w, ln_b, ws);
  hipLaunchKernelGGL(fold_fc_kernel, dim3(2), dim3(256), 0, stream, fc_w, ws);
  hipLaunchKernelGGL(fused_main_kernel, dim3(NB * (NHW / 256)), dim3(256), 0,
                     stream, x, fc_b, ws, out);
}